// IMUPreintegrator_4114578669830
// MI455X (gfx1250) — compile-verified
//
#include <hip/hip_runtime.h>
#include <math.h>

#define B_ 16
#define F_ 4096
#define C_ 64              // chunks per batch
#define CF_ (F_ / C_)      // frames per chunk = 64
#define LT 17              // padded LDS tile stride (conflict-free)
#define GRAV 9.81007f
#define ANGCOV (0.00016968f * 0.00016968f)
#define ACCCOV (0.002f * 0.002f)
#define EPS_ 1e-8f

typedef __attribute__((ext_vector_type(2))) float v2f;
typedef __attribute__((ext_vector_type(8))) float v8f;

// ---------------- quaternion helpers ----------------
__device__ __forceinline__ float4 qmul4(float4 a, float4 b) {
    float4 r;
    r.x = a.w * b.x + b.w * a.x + (a.y * b.z - a.z * b.y);
    r.y = a.w * b.y + b.w * a.y + (a.z * b.x - a.x * b.z);
    r.z = a.w * b.z + b.w * a.z + (a.x * b.y - a.y * b.x);
    r.w = a.w * b.w - (a.x * b.x + a.y * b.y + a.z * b.z);
    return r;
}
__device__ __forceinline__ float3 qrot3(float4 q, float3 v) {
    float tx = 2.f * (q.y * v.z - q.z * v.y);
    float ty = 2.f * (q.z * v.x - q.x * v.z);
    float tz = 2.f * (q.x * v.y - q.y * v.x);
    float3 r;
    r.x = v.x + q.w * tx + (q.y * tz - q.z * ty);
    r.y = v.y + q.w * ty + (q.z * tx - q.x * tz);
    r.z = v.z + q.w * tz + (q.x * ty - q.y * tx);
    return r;
}
__device__ __forceinline__ float4 qinv4(float4 q) { return make_float4(-q.x, -q.y, -q.z, q.w); }

__device__ __forceinline__ void qmat3(float4 q, float R[9]) {
    float x = q.x, y = q.y, z = q.z, w = q.w;
    R[0] = 1.f - 2.f * (y * y + z * z); R[1] = 2.f * (x * y - w * z); R[2] = 2.f * (x * z + w * y);
    R[3] = 2.f * (x * y + w * z); R[4] = 1.f - 2.f * (x * x + z * z); R[5] = 2.f * (y * z - w * x);
    R[6] = 2.f * (x * z - w * y); R[7] = 2.f * (y * z + w * x); R[8] = 1.f - 2.f * (x * x + y * y);
}

// ---------------- WMMA 16x16 fp32 tile ops (V_WMMA_F32_16X16X4_F32) ----------------
// A 16x4 f32: lanes 0-15 hold M=lane {K=k0,k0+1}, lanes 16-31 hold M=lane-16 {K=k0+2,k0+3}
// B 4x16 f32: lanes 0-15 hold N=lane rows {k0,k0+1}, lanes 16-31 rows {k0+2,k0+3}
// C/D 16x16 : VGPR r -> row r (lanes 0-15) / row r+8 (lanes 16-31), col = lane&15
__device__ __forceinline__ v8f wm_mul(const float* X, const float* Y, v8f acc) {
    const int lane = threadIdx.x & 31;
    const int m = lane & 15;
    const int hk = (lane >> 4) * 2;
#pragma unroll
    for (int kc = 0; kc < 4; ++kc) {
        const int k0 = kc * 4 + hk;
        v2f a; a.x = X[m * LT + k0]; a.y = X[m * LT + k0 + 1];
        v2f b; b.x = Y[k0 * LT + m]; b.y = Y[(k0 + 1) * LT + m];
        acc = __builtin_amdgcn_wmma_f32_16x16x4_f32(false, a, false, b, (short)0, acc, false, false);
    }
    return acc;
}
// D = X @ Y^T
__device__ __forceinline__ v8f wm_mul_bt(const float* X, const float* Y, v8f acc) {
    const int lane = threadIdx.x & 31;
    const int m = lane & 15;
    const int hk = (lane >> 4) * 2;
#pragma unroll
    for (int kc = 0; kc < 4; ++kc) {
        const int k0 = kc * 4 + hk;
        v2f a; a.x = X[m * LT + k0]; a.y = X[m * LT + k0 + 1];
        v2f b; b.x = Y[m * LT + k0]; b.y = Y[m * LT + k0 + 1];
        acc = __builtin_amdgcn_wmma_f32_16x16x4_f32(false, a, false, b, (short)0, acc, false, false);
    }
    return acc;
}
__device__ __forceinline__ void wm_store(float* T, v8f c) {
    const int lane = threadIdx.x & 31;
    const int n = lane & 15;
    const int rb = (lane >> 4) * 8;
#pragma unroll
    for (int r = 0; r < 8; ++r) T[(rb + r) * LT + n] = c[r];
}
__device__ __forceinline__ void tile_to_g(float* g, const float* T) {
    const int lane = threadIdx.x & 31;
    for (int idx = lane; idx < 256; idx += 32) g[idx] = T[(idx >> 4) * LT + (idx & 15)];
}
__device__ __forceinline__ void g_to_tile(float* T, const float* g) {
    const int lane = threadIdx.x & 31;
    for (int idx = lane; idx < 256; idx += 32) T[(idx >> 4) * LT + (idx & 15)] = g[idx];
}
__device__ __forceinline__ void set_i9(float* T) {
    const int lane = threadIdx.x & 31;
    for (int idx = lane; idx < 16 * LT; idx += 32) T[idx] = 0.f;
    if (lane < 9) T[lane * LT + lane] = 1.f;
}

// ---------------- A and Q (process-noise) tile builders ----------------
__device__ __forceinline__ void build_A(float* T, float4 e, float4 ri, float3 av, float d) {
    float dR[9], Rm[9];
    qmat3(e, dR); qmat3(ri, Rm);
    const float S[9] = {0.f, -av.z, av.y, av.z, 0.f, -av.x, -av.y, av.x, 0.f};
    float RH[9];
#pragma unroll
    for (int i = 0; i < 3; ++i)
#pragma unroll
        for (int j = 0; j < 3; ++j)
            RH[i * 3 + j] = Rm[i * 3] * S[j] + Rm[i * 3 + 1] * S[3 + j] + Rm[i * 3 + 2] * S[6 + j];
    const int lane = threadIdx.x & 31;
    for (int idx = lane; idx < 16 * LT; idx += 32) T[idx] = 0.f;
    for (int idx = lane; idx < 81; idx += 32) {
        int r = idx / 9, c = idx % 9;
        int br = r / 3, bc = c / 3, rr = r % 3, cc = c % 3;
        float v = 0.f;
        if (br == 0 && bc == 0) v = dR[cc * 3 + rr];                 // dRm^T
        else if (br == 1 && bc == 0) v = -RH[rr * 3 + cc] * d;
        else if (br == 1 && bc == 1) v = (rr == cc) ? 1.f : 0.f;
        else if (br == 2 && bc == 0) v = -0.5f * RH[rr * 3 + cc] * d * d;
        else if (br == 2 && bc == 1) v = (rr == cc) ? d : 0.f;
        else if (br == 2 && bc == 2) v = (rr == cc) ? 1.f : 0.f;
        T[r * LT + c] = v;
    }
}

__device__ __forceinline__ float dotrow(const float* A, int i, const float* Bm, int j) {
    return A[i * 3] * Bm[j * 3] + A[i * 3 + 1] * Bm[j * 3 + 1] + A[i * 3 + 2] * Bm[j * 3 + 2];
}

__device__ __forceinline__ void build_Q(float* T, float4 e, float4 ri, float d) {
    // phi = qlog(e)
    float n2 = e.x * e.x + e.y * e.y + e.z * e.z;
    bool sm = n2 < EPS_;
    float n = sqrtf(sm ? 1.f : n2);
    float k = sm ? (2.f / e.w) * (1.f - n2 / (3.f * e.w * e.w)) : 2.f * atan2f(n, e.w) / n;
    float px = k * e.x, py = k * e.y, pz = k * e.z;
    // Jr(phi)
    float t2 = px * px + py * py + pz * pz;
    bool sm2 = t2 < EPS_;
    float th = sqrtf(sm2 ? 1.f : t2);
    float c1 = sm2 ? (0.5f - t2 / 24.f) : ((1.f - cosf(th)) / t2);
    float c2 = sm2 ? (1.f / 6.f - t2 / 120.f) : ((th - sinf(th)) / (t2 * th));
    const float K[9] = {0.f, -pz, py, pz, 0.f, -px, -py, px, 0.f};
    float KK[9];
#pragma unroll
    for (int i = 0; i < 3; ++i)
#pragma unroll
        for (int j = 0; j < 3; ++j)
            KK[i * 3 + j] = K[i * 3] * K[j] + K[i * 3 + 1] * K[3 + j] + K[i * 3 + 2] * K[6 + j];
    float Jr[9];
#pragma unroll
    for (int i = 0; i < 9; ++i) Jr[i] = ((i % 4 == 0) ? 1.f : 0.f) - c1 * K[i] + c2 * KK[i];
    float dR[9], Rm[9];
    qmat3(e, dR); qmat3(ri, Rm);
    const float cg = ANGCOV * d;
    const float ca = ACCCOV * d;
    const float cx = ACCCOV * 0.5f * d * d;
    const float cd = ACCCOV * 0.25f * d * d * d;
    const int lane = threadIdx.x & 31;
    for (int idx = lane; idx < 16 * LT; idx += 32) T[idx] = 0.f;
    for (int idx = lane; idx < 81; idx += 32) {
        int r = idx / 9, c = idx % 9;
        int br = r / 3, bc = c / 3, rr = r % 3, cc = c % 3;
        float v = 0.f;
        if (br == 0 && bc == 0) v = cg * dotrow(Jr, rr, Jr, cc);
        else if (br == 1 && bc == 1) v = ca * dotrow(Rm, rr, Rm, cc);
        else if (br == 1 && bc == 2) v = cx * dotrow(Rm, rr, dR, cc);
        else if (br == 2 && bc == 1) v = cx * dotrow(Rm, cc, dR, rr);
        else if (br == 2 && bc == 2) v = cd * dotrow(dR, rr, dR, cc);
        T[r * LT + c] = v;
    }
}

// ---------------- K1: dr = so3_exp(ang*dt) ----------------
__global__ void k_dr(const float* __restrict__ dt, const float* __restrict__ ang,
                     float* __restrict__ e) {
    int i = blockIdx.x * blockDim.x + threadIdx.x;
    if (i >= B_ * F_) return;
    float d = dt[i];
    float px = ang[i * 3] * d, py = ang[i * 3 + 1] * d, pz = ang[i * 3 + 2] * d;
    float t2 = px * px + py * py + pz * pz;
    bool sm = t2 < EPS_;
    float th = sqrtf(sm ? 1.f : t2);
    float s = sm ? (0.5f - t2 * (1.f / 48.f)) : (sinf(0.5f * th) / th);
    float w = sm ? (1.f - t2 * 0.125f) : cosf(0.5f * th);
    float4 q = make_float4(s * px, s * py, s * pz, w);
    *(float4*)(e + (size_t)i * 4) = q;
}

// ---------------- K2: per-batch inclusive quaternion scan ----------------
__global__ __launch_bounds__(1024) void k_qscan(const float* __restrict__ e,
                                                float* __restrict__ sI) {
    __shared__ float4 lds[1024];
    const int b = blockIdx.x, t = threadIdx.x;
    const float* eb = e + (size_t)b * F_ * 4;
    float4 loc[4], a;
#pragma unroll
    for (int k = 0; k < 4; ++k) {
        float4 q = *(const float4*)(eb + (size_t)(t * 4 + k) * 4);
        a = (k == 0) ? q : qmul4(a, q);
        loc[k] = a;
    }
    lds[t] = a;
    for (int off = 1; off < 1024; off <<= 1) {
        __syncthreads();
        float4 v = lds[t];
        float4 nv = (t >= off) ? qmul4(lds[t - off], v) : v;
        __syncthreads();
        lds[t] = nv;
    }
    __syncthreads();
    float4 pre = (t == 0) ? make_float4(0.f, 0.f, 0.f, 1.f) : lds[t - 1];
    float* ob = sI + (size_t)b * F_ * 4;
#pragma unroll
    for (int k = 0; k < 4; ++k) {
        float4 r = (t == 0) ? loc[k] : qmul4(pre, loc[k]);
        *(float4*)(ob + (size_t)(t * 4 + k) * 4) = r;
    }
}

// ---------------- block scans for vec3 / scalar ----------------
__device__ __forceinline__ void scan3(float* lx, float* ly, float* lz, int t, float3 loc[4]) {
    float ax = 0.f, ay = 0.f, az = 0.f;
#pragma unroll
    for (int k = 0; k < 4; ++k) {
        ax += loc[k].x; ay += loc[k].y; az += loc[k].z;
        loc[k].x = ax; loc[k].y = ay; loc[k].z = az;
    }
    __syncthreads();
    lx[t] = ax; ly[t] = ay; lz[t] = az;
    for (int off = 1; off < 1024; off <<= 1) {
        __syncthreads();
        float vx = lx[t], vy = ly[t], vz = lz[t];
        if (t >= off) { vx += lx[t - off]; vy += ly[t - off]; vz += lz[t - off]; }
        __syncthreads();
        lx[t] = vx; ly[t] = vy; lz[t] = vz;
    }
    __syncthreads();
    if (t > 0) {
        float pxx = lx[t - 1], pyy = ly[t - 1], pzz = lz[t - 1];
#pragma unroll
        for (int k = 0; k < 4; ++k) { loc[k].x += pxx; loc[k].y += pyy; loc[k].z += pzz; }
    }
}
__device__ __forceinline__ void scan1(float* l, int t, float loc[4]) {
    float a = 0.f;
#pragma unroll
    for (int k = 0; k < 4; ++k) { a += loc[k]; loc[k] = a; }
    __syncthreads();
    l[t] = a;
    for (int off = 1; off < 1024; off <<= 1) {
        __syncthreads();
        float v = l[t];
        if (t >= off) v += l[t - off];
        __syncthreads();
        l[t] = v;
    }
    __syncthreads();
    if (t > 0) {
        float p = l[t - 1];
#pragma unroll
        for (int k = 0; k < 4; ++k) loc[k] += p;
    }
}

// ---------------- K3: vector integrations + outputs ----------------
__global__ __launch_bounds__(1024) void k_vec(const float* __restrict__ dt,
                                              const float* __restrict__ acc,
                                              const float* __restrict__ sI,
                                              const float* __restrict__ ip,
                                              const float* __restrict__ iq,
                                              const float* __restrict__ iv,
                                              float* __restrict__ rot_p,
                                              float* __restrict__ vel_p,
                                              float* __restrict__ pos_p) {
    __shared__ float lx[1024], ly[1024], lz[1024];
    const int b = blockIdx.x, t = threadIdx.x;
    const float4 qi = make_float4(iq[0], iq[1], iq[2], iq[3]);
    const float3 v0 = make_float3(iv[0], iv[1], iv[2]);
    const float3 p0 = make_float3(ip[0], ip[1], ip[2]);
    const float3 g = make_float3(0.f, 0.f, GRAV);
    float dts[4];
    float3 ra[4], s1[4];
#pragma unroll
    for (int k = 0; k < 4; ++k) {
        int f = t * 4 + k;
        size_t gi = (size_t)b * F_ + f;
        float d = dt[gi]; dts[k] = d;
        float4 ri = *(const float4*)(sI + gi * 4);
        float4 ir = qmul4(qi, ri);
        *(float4*)(rot_p + gi * 4) = ir;
        float3 av = make_float3(acc[gi * 3], acc[gi * 3 + 1], acc[gi * 3 + 2]);
        float3 gb = qrot3(qinv4(ir), g);
        float3 aa = make_float3(av.x - gb.x, av.y - gb.y, av.z - gb.z);
        float4 pr = (f == 0) ? make_float4(0.f, 0.f, 0.f, 1.f)
                             : *(const float4*)(sI + ((size_t)b * F_ + f - 1) * 4);
        ra[k] = qrot3(pr, aa);
        s1[k] = make_float3(ra[k].x * d, ra[k].y * d, ra[k].z * d);
    }
    scan3(lx, ly, lz, t, s1);   // s1 = inclusive cumsum(ra*dt) == incre_v[f+1]
    float3 dp[4];
#pragma unroll
    for (int k = 0; k < 4; ++k) {
        int f = t * 4 + k;
        size_t gi = (size_t)b * F_ + f;
        float3 vr = qrot3(qi, s1[k]);
        vel_p[gi * 3] = v0.x + vr.x; vel_p[gi * 3 + 1] = v0.y + vr.y; vel_p[gi * 3 + 2] = v0.z + vr.z;
        float d = dts[k];
        // exclusive prefix * dt + 0.5*ra*dt^2
        dp[k].x = (s1[k].x - ra[k].x * d) * d + 0.5f * ra[k].x * d * d;
        dp[k].y = (s1[k].y - ra[k].y * d) * d + 0.5f * ra[k].y * d * d;
        dp[k].z = (s1[k].z - ra[k].z * d) * d + 0.5f * ra[k].z * d * d;
    }
    scan3(lx, ly, lz, t, dp);   // dp = incre_p[f+1]
    float st[4];
#pragma unroll
    for (int k = 0; k < 4; ++k) st[k] = dts[k];
    scan1(lx, t, st);           // st = incre_t[f+1]
#pragma unroll
    for (int k = 0; k < 4; ++k) {
        size_t gi = (size_t)b * F_ + t * 4 + k;
        float3 prot = qrot3(qi, dp[k]);
        pos_p[gi * 3]     = p0.x + prot.x + v0.x * st[k];
        pos_p[gi * 3 + 1] = p0.y + prot.y + v0.y * st[k];
        pos_p[gi * 3 + 2] = p0.z + prot.z + v0.z * st[k];
    }
}

// ---------------- K4a: per-chunk product of A matrices (WMMA) ----------------
__global__ __launch_bounds__(32) void k_chunkP(const float* __restrict__ e,
                                               const float* __restrict__ sI,
                                               const float* __restrict__ acc,
                                               const float* __restrict__ dt,
                                               float* __restrict__ wsP) {
    __shared__ float tP[16 * LT], tA[16 * LT];
    const int bc = blockIdx.x, b = bc / C_, c = bc % C_;
    set_i9(tP);
    __syncthreads();
    const int f0 = c * CF_;
    for (int k = 0; k < CF_; ++k) {
        const size_t gi = (size_t)b * F_ + (f0 + k);
        float4 ef = *(const float4*)(e + gi * 4);
        float4 ri = *(const float4*)(sI + gi * 4);
        float3 av = make_float3(acc[gi * 3], acc[gi * 3 + 1], acc[gi * 3 + 2]);
        float d = dt[gi];
        build_A(tA, ef, ri, av, d);
        __syncthreads();
        v8f z = {};
        v8f r = wm_mul(tP, tA, z);   // P = P @ A[f]
        __syncthreads();
        wm_store(tP, r);
        __syncthreads();
    }
    tile_to_g(wsP + ((size_t)b * C_ + c) * 256, tP);
}

// ---------------- K4b: per-batch suffix products over chunks ----------------
__global__ __launch_bounds__(32) void k_suffix(const float* __restrict__ wsP,
                                               float* __restrict__ wsR,
                                               float* __restrict__ wsCov) {
    __shared__ float tR[16 * LT], tP[16 * LT];
    const int b = blockIdx.x;
    const int lane = threadIdx.x & 31;
    for (int idx = lane; idx < 81; idx += 32) wsCov[b * 81 + idx] = 0.f;
    set_i9(tR);
    __syncthreads();
    for (int c = C_ - 1; c >= 0; --c) {
        tile_to_g(wsR + ((size_t)b * (C_ + 1) + c) * 256, tR);   // R_c
        g_to_tile(tP, wsP + ((size_t)b * C_ + c) * 256);
        __syncthreads();
        v8f z = {};
        v8f r = wm_mul(tP, tR, z);   // R_{c-1} = P_c @ R_c
        __syncthreads();
        wm_store(tR, r);
        __syncthreads();
    }
    tile_to_g(wsR + ((size_t)b * (C_ + 1) + C_) * 256, tR);      // full product A[0..F-1]
}

// ---------------- K4c: per-chunk covariance accumulation (WMMA) ----------------
__global__ __launch_bounds__(32) void k_cov(const float* __restrict__ e,
                                            const float* __restrict__ sI,
                                            const float* __restrict__ acc,
                                            const float* __restrict__ dt,
                                            const float* __restrict__ wsR,
                                            float* __restrict__ wsCov) {
    __shared__ float tR[16 * LT], tN[16 * LT], tQ[16 * LT], tA[16 * LT], tT1[16 * LT], tT2[16 * LT];
    const int bc = blockIdx.x, b = bc / C_, c = bc % C_;
    g_to_tile(tR, wsR + ((size_t)b * (C_ + 1) + c) * 256);
    set_i9(tN);
    __syncthreads();
    v8f S = {};
    const int f1 = c * CF_ + CF_ - 1;
    for (int k = 0; k < CF_; ++k) {
        const int f = f1 - k;
        const size_t gi = (size_t)b * F_ + f;
        float4 ef = *(const float4*)(e + gi * 4);
        float4 ri = *(const float4*)(sI + gi * 4);
        float d = dt[gi];
        build_Q(tQ, ef, ri, d);
        __syncthreads();
        v8f z0 = {};
        v8f a1 = wm_mul(tR, tQ, z0);          __syncthreads(); wm_store(tT1, a1); __syncthreads();
        v8f z1 = {};
        v8f a2 = wm_mul_bt(tT1, tR, z1);      __syncthreads(); wm_store(tT2, a2); __syncthreads();
        v8f z2 = {};
        v8f a3 = wm_mul(tN, tT2, z2);         __syncthreads(); wm_store(tT1, a3); __syncthreads();
        S = wm_mul_bt(tT1, tN, S);            // cov += Nl (R Q R^T) Nl^T
        float3 av = make_float3(acc[gi * 3], acc[gi * 3 + 1], acc[gi * 3 + 2]);
        build_A(tA, ef, ri, av, d);
        __syncthreads();
        v8f z3 = {};
        v8f a4 = wm_mul(tA, tN, z3);          // Nl = A[f] @ Nl
        __syncthreads();
        wm_store(tN, a4);
        __syncthreads();
    }
    const int lane = threadIdx.x & 31;
    const int n = lane & 15;
    const int rb = (lane >> 4) * 8;
    if (n < 9) {
#pragma unroll
        for (int r = 0; r < 8; ++r) {
            int row = rb + r;
            if (row < 9) atomicAdd(&wsCov[b * 81 + row * 9 + n], S[r]);
        }
    }
}

// ---------------- K5: add init_cov term, write cov ----------------
__global__ void k_final(const float* __restrict__ wsCov, const float* __restrict__ wsR,
                        const float* __restrict__ init_cov, float* __restrict__ cov_o) {
    const int b = blockIdx.x, t = threadIdx.x;
    if (t >= 81) return;
    const int i = t / 9, j = t % 9;
    const float* M = wsR + ((size_t)b * (C_ + 1) + C_) * 256;   // 16x16 row-major
    const float* C0 = init_cov + b * 81;
    float a = wsCov[b * 81 + t];
    for (int k = 0; k < 9; ++k) {
        float mik = M[i * 16 + k];
        float s = 0.f;
        for (int l = 0; l < 9; ++l) s += C0[k * 9 + l] * M[j * 16 + l];
        a += mik * s;
    }
    cov_o[b * 81 + t] = a;
}

// ---------------- launch ----------------
extern "C" void kernel_launch(void* const* d_in, const int* in_sizes, int n_in,
                              void* d_out, int out_size, void* d_ws, size_t ws_size,
                              hipStream_t stream) {
    (void)in_sizes; (void)n_in; (void)out_size; (void)ws_size;
    const float* dt  = (const float*)d_in[0];
    const float* ang = (const float*)d_in[1];
    const float* acc = (const float*)d_in[2];
    const float* ip  = (const float*)d_in[3];
    const float* iq  = (const float*)d_in[4];
    const float* iv  = (const float*)d_in[5];
    const float* ic  = (const float*)d_in[6];

    float* out = (float*)d_out;
    float* rot_p = out;
    float* vel_p = out + (size_t)B_ * F_ * 4;
    float* pos_p = vel_p + (size_t)B_ * F_ * 3;
    float* cov_o = pos_p + (size_t)B_ * F_ * 3;

    float* ws    = (float*)d_ws;
    float* ws_e  = ws;                                   // B*F*4
    float* ws_s  = ws_e + (size_t)B_ * F_ * 4;           // B*F*4
    float* ws_P  = ws_s + (size_t)B_ * F_ * 4;           // B*C*256
    float* ws_R  = ws_P + (size_t)B_ * C_ * 256;         // B*(C+1)*256
    float* ws_cv = ws_R + (size_t)B_ * (C_ + 1) * 256;   // B*81

    k_dr<<<(B_ * F_ + 255) / 256, 256, 0, stream>>>(dt, ang, ws_e);
    k_qscan<<<B_, 1024, 0, stream>>>(ws_e, ws_s);
    k_vec<<<B_, 1024, 0, stream>>>(dt, acc, ws_s, ip, iq, iv, rot_p, vel_p, pos_p);
    k_chunkP<<<B_ * C_, 32, 0, stream>>>(ws_e, ws_s, acc, dt, ws_P);
    k_suffix<<<B_, 32, 0, stream>>>(ws_P, ws_R, ws_cv);
    k_cov<<<B_ * C_, 32, 0, stream>>>(ws_e, ws_s, acc, dt, ws_R, ws_cv);
    k_final<<<B_, 96, 0, stream>>>(ws_cv, ws_R, ic, cov_o);
}